// SpatialConsistencyLoss_30588757082425
// MI455X (gfx1250) — compile-verified
//
#include <hip/hip_runtime.h>

typedef float v2f __attribute__((ext_vector_type(2)));
typedef float v8f __attribute__((ext_vector_type(8)));

#define HW 65536   // 256*256
#define NB 32
#define NP 8

// ---------------------------------------------------------------- zero ws
__global__ void k_zero(int* __restrict__ ws, int n) {
    int i = blockIdx.x * blockDim.x + threadIdx.x;
    if (i < n) ws[i] = 0;
}

// ------------------------------------------------- mask centroids (b,p)
// One block per (b, part). Channels 1..8 of the 9-channel mask tensor.
__global__ void k_mask_centers(const float* __restrict__ masks,
                               int* __restrict__ out /* 32*8*3 ints */) {
    int bp = blockIdx.x;                // 0..255
    int b = bp >> 3, p = bp & 7;
    const float4* src = (const float4*)(masks + ((size_t)b * 9 + 1 + p) * HW);
    int tid = threadIdx.x;

    int cnt = 0, sx = 0, sy = 0;
    #pragma unroll 4
    for (int i = 0; i < 64; ++i) {
        int idx4 = i * 256 + tid;       // coalesced b128 loads
        float4 v = src[idx4];
        int pix = idx4 * 4;
        float vv[4] = {v.x, v.y, v.z, v.w};
        #pragma unroll
        for (int c = 0; c < 4; ++c) {
            if (vv[c] > 0.5f) {
                int pc = pix + c;
                cnt += 1; sx += (pc >> 8); sy += (pc & 255);
            }
        }
    }
    // wave32 reduce
    for (int m = 16; m; m >>= 1) {
        cnt += __shfl_xor(cnt, m, 32);
        sx  += __shfl_xor(sx,  m, 32);
        sy  += __shfl_xor(sy,  m, 32);
    }
    __shared__ int acc[3];
    if (tid < 3) acc[tid] = 0;
    __syncthreads();
    if ((tid & 31) == 0) {
        atomicAdd(&acc[0], cnt);
        atomicAdd(&acc[1], sx);
        atomicAdd(&acc[2], sy);
    }
    __syncthreads();
    if (tid < 3) out[bp * 3 + tid] = acc[tid];
}

// -------------------------------- keypoint part centroids via f32 WMMA
// part_hm(8 x HW) = AGG(8x23) * K(23 x HW), done as 16x16x4 f32 WMMAs:
// A = AGG padded to 16x24 (zero rows 8..15, zero col 23),
// B = 16-pixel slab of keypoint channels; 6 chained K-tiles per slab.
__global__ void k_kp_centers(const float* __restrict__ kp,
                             int* __restrict__ out /* 32*8*3 ints */) {
    const unsigned tabv[8] = {0x1Fu,     0x1860u,  0x2A0u,   0x540u,
                              0xA800u,   0x15000u, 0xE8000u, 0x710000u};
    int blk  = blockIdx.x;              // 0..255
    int b    = blk >> 3;                // batch
    int bi   = blk & 7;                 // chunk within batch
    const float* src = kp + (size_t)b * 23 * HW;

    int tid  = threadIdx.x;
    int lane = tid & 31, wave = tid >> 5;
    int hi   = lane >> 4;               // half-wave selector
    int nl   = lane & 15;               // N index (pixel) / M index (A row)

    // A-matrix tiles (16x4 f32 layout: lanes 0-15 -> K=0,1; lanes 16-31 -> K=2,3)
    v2f A[6];
    #pragma unroll
    for (int t = 0; t < 6; ++t) {
        #pragma unroll
        for (int j = 0; j < 2; ++j) {
            int k = 4 * t + 2 * hi + j;
            unsigned bit = (nl < 8 && k < 23) ? ((tabv[nl & 7] >> k) & 1u) : 0u;
            A[t][j] = (float)bit;
        }
    }
    // Clamped channel indices for B loads: A columns k>=23 are zero, so any
    // loaded value there is annihilated by the multiply (no EXEC divergence).
    int kc[12];
    #pragma unroll
    for (int t = 0; t < 6; ++t)
        #pragma unroll
        for (int j = 0; j < 2; ++j) {
            int k = 4 * t + 2 * hi + j;
            kc[2 * t + j] = (k < 23) ? k : 22;
        }

    int cnt[8] = {0}, sx[8] = {0}, sy[8] = {0};
    int tile0 = bi * 512 + wave;        // 512 tiles per block, 8 waves
    for (int i = 0; i < 64; ++i) {
        int tile = tile0 + 8 * i;
        int px   = tile * 16;
        int pr   = (px + nl) >> 8;      // row of this lane's pixel
        int pc   = (px + nl) & 255;     // col of this lane's pixel
        v8f C = {};
        #pragma unroll
        for (int t = 0; t < 6; ++t) {
            v2f B;
            B[0] = src[(size_t)kc[2 * t]     * HW + px + nl];
            B[1] = src[(size_t)kc[2 * t + 1] * HW + px + nl];
            C = __builtin_amdgcn_wmma_f32_16x16x4_f32(
                    false, A[t], false, B, (short)0, C, false, false);
        }
        // VGPR p of C: lanes 0-15 = part p (lanes 16-31 = zero rows -> inactive)
        #pragma unroll
        for (int p = 0; p < 8; ++p) {
            bool act = C[p] > 0.3f;
            cnt[p] += act ? 1  : 0;
            sx[p]  += act ? pr : 0;
            sy[p]  += act ? pc : 0;
        }
    }

    __shared__ int acc[24];
    if (tid < 24) acc[tid] = 0;
    __syncthreads();
    #pragma unroll
    for (int p = 0; p < 8; ++p) {
        int c = cnt[p], x = sx[p], y = sy[p];
        for (int m = 16; m; m >>= 1) {
            c += __shfl_xor(c, m, 32);
            x += __shfl_xor(x, m, 32);
            y += __shfl_xor(y, m, 32);
        }
        if (lane == 0) {
            atomicAdd(&acc[p * 3 + 0], c);
            atomicAdd(&acc[p * 3 + 1], x);
            atomicAdd(&acc[p * 3 + 2], y);
        }
    }
    __syncthreads();
    if (tid < 24) atomicAdd(&out[b * 24 + tid], acc[tid]);
}

// ------------------------------------------------------- final scalar
__global__ void k_final(const int* __restrict__ msk,
                        const int* __restrict__ kpt,
                        float* __restrict__ out) {
    int tid = threadIdx.x;              // one (b,p) entry per thread
    const int* mt = msk + tid * 3;
    const int* kt = kpt + tid * 3;

    auto center = [](int s, int c) -> float {
        float mx = (float)s / fmaxf((float)c, 1.0f);
        return (c > 0 && mx > 0.0f) ? mx : 0.0f;
    };
    float mcx = center(mt[1], mt[0]);
    float mcy = center(mt[2], mt[0]);
    float kcx = center(kt[1], kt[0]);
    float kcy = center(kt[2], kt[0]);

    bool code = (kcx == 0.0f) || (kcy == 0.0f) || (mcx == 0.0f) || (mcy == 0.0f);
    float dx = mcx - kcx, dy = mcy - kcy;
    float num = code ? 0.0f : (dx * dx + dy * dy);
    float den = code ? 0.0f : 2.0f;

    __shared__ float sn[256], sd[256];
    sn[tid] = num; sd[tid] = den;
    __syncthreads();
    for (int s = 128; s; s >>= 1) {
        if (tid < s) { sn[tid] += sn[tid + s]; sd[tid] += sd[tid + s]; }
        __syncthreads();
    }
    if (tid == 0) out[0] = 1e-5f * (sn[0] / sd[0]);
}

extern "C" void kernel_launch(void* const* d_in, const int* in_sizes, int n_in,
                              void* d_out, int out_size, void* d_ws, size_t ws_size,
                              hipStream_t stream) {
    const float* masks = (const float*)d_in[0];   // (32, 9, 256, 256) f32
    const float* kps   = (const float*)d_in[1];   // (32, 23, 256, 256) f32
    float* out = (float*)d_out;
    int* ws    = (int*)d_ws;
    int* wsm   = ws;         // 768 ints: mask (cnt,sx,sy) per (b,p)
    int* wsk   = ws + 768;   // 768 ints: keypoint (cnt,sx,sy) per (b,p)

    k_zero        <<<6,   256, 0, stream>>>(ws, 1536);
    k_mask_centers<<<256, 256, 0, stream>>>(masks, wsm);
    k_kp_centers  <<<256, 256, 0, stream>>>(kps, wsk);
    k_final       <<<1,   256, 0, stream>>>(wsm, wsk, out);
}